// GCNMultiLabel_13915694039845
// MI455X (gfx1250) — compile-verified
//
#include <hip/hip_runtime.h>

typedef float v2f __attribute__((ext_vector_type(2)));
typedef float v8f __attribute__((ext_vector_type(8)));

#define DIN  128
#define HID  128
#define D2   64
#define NOUT 1000

// ---------------------------------------------------------------------------
// Utility kernels
// ---------------------------------------------------------------------------
__global__ void fill_zero(float* __restrict__ p, long long n) {
    long long i = (long long)blockIdx.x * blockDim.x + threadIdx.x;
    if (i < n) p[i] = 0.0f;
}

__global__ void deg_count(const long long* __restrict__ dst,
                          float* __restrict__ deg, int E) {
    int i = blockIdx.x * blockDim.x + threadIdx.x;
    if (i < E) atomicAdd(&deg[(int)dst[i]], 1.0f);
}

// deg -> dinv = rsqrt(deg + 1)   (in place)
__global__ void deg_to_dinv(float* __restrict__ deg, int n) {
    int i = blockIdx.x * blockDim.x + threadIdx.x;
    if (i < n) deg[i] = rsqrtf(deg[i] + 1.0f);
}

// ---------------------------------------------------------------------------
// Edge aggregation: agg[dst] += h[src] * dinv[src]*dinv[dst]
// One thread handles 4 consecutive features of one edge (float4 gather,
// 4 global_atomic_add_f32 scatters).
// ---------------------------------------------------------------------------
__global__ void edge_aggregate(const float* __restrict__ H,
                               const long long* __restrict__ src,
                               const long long* __restrict__ dst,
                               const float* __restrict__ dinv,
                               float* __restrict__ agg, int E, int F) {
    long long t = (long long)blockIdx.x * blockDim.x + threadIdx.x;
    const int cpe = F >> 2;                      // float4 chunks per edge
    if (t >= (long long)E * cpe) return;
    int e = (int)(t / cpe);
    int c = (int)(t % cpe);
    int s = (int)src[e];
    int d = (int)dst[e];
    float norm = dinv[s] * dinv[d];
    const float4 h = *((const float4*)(H + (long long)s * F) + c);
    float* o = agg + (long long)d * F + (long long)c * 4;
    atomicAdd(o + 0, h.x * norm);
    atomicAdd(o + 1, h.y * norm);
    atomicAdd(o + 2, h.z * norm);
    atomicAdd(o + 3, h.w * norm);
}

// agg = relu(agg + H*dinv^2 + b)   (in place on agg)
__global__ void self_bias_relu(float* __restrict__ agg,
                               const float* __restrict__ H,
                               const float* __restrict__ dinv,
                               const float* __restrict__ b,
                               int n, int F) {
    long long t = (long long)blockIdx.x * blockDim.x + threadIdx.x;
    if (t >= (long long)n * F) return;
    int row = (int)(t / F);
    int f   = (int)(t % F);
    float di = dinv[row];
    float v = agg[t] + H[t] * di * di + b[f];
    agg[t] = v > 0.0f ? v : 0.0f;
}

// ---------------------------------------------------------------------------
// fp32 WMMA GEMM:  C[M,N] = A[M,K] @ B[K,N] (+ bias[N])
//
// Block = 8 waves = 8 consecutive 16-row m-tiles x one 64-column n-group.
// The 64-column B panel is staged TRANSPOSED into LDS (Bt[col][k], row
// stride K+2 floats -> 8-byte aligned v2f reads, no heavy bank conflicts),
// shared by all 8 waves.
//
// WMMA fragment layouts (V_WMMA_F32_16X16X4_F32, wave32):
//   A 16x4 : lane L(0..15) -> M=L, holds K=k0,k0+1 ; lane 16+L -> K=k0+2,k0+3
//   B 4x16 : lane L(0..15) -> N=L, rows k0,k0+1    ; lane 16+L -> rows k0+2,k0+3
//   D 16x16: vgpr r, lanes 0-15 -> M=r ; lanes 16-31 -> M=8+r
//
// Requires M % 16 == 0 (true here: M = 100000). NGUARD=true handles a
// partial column group (FC tail, cols 960..999).
// ---------------------------------------------------------------------------
template <int K, bool BIAS, bool NGUARD>
__global__ __launch_bounds__(256)
void wmma_gemm_f32(const float* __restrict__ A, const float* __restrict__ B,
                   const float* __restrict__ bias, float* __restrict__ C,
                   int M, int N, int n0base) {
    constexpr int NCOLS = 64;                 // columns per block
    constexpr int LDB   = K + 2;              // LDS row stride (floats)
    __shared__ float Bt[NCOLS * LDB];

    const int n0 = n0base + blockIdx.y * NCOLS;

    // ---- stage B panel (transposed) into LDS, coalesced global reads ----
    for (int idx = threadIdx.x; idx < K * NCOLS; idx += 256) {
        int k = idx >> 6;                     // 0..K-1
        int c = idx & (NCOLS - 1);            // 0..63
        int g = n0 + c;
        float v;
        if (NGUARD) v = (g < N) ? B[(long long)k * N + g] : 0.0f;
        else        v = B[(long long)k * N + g];
        Bt[c * LDB + k] = v;
    }
    __syncthreads();

    const int wave  = threadIdx.x >> 5;
    const int lane  = threadIdx.x & 31;
    const int l15   = lane & 15;
    const int khalf = lane >> 4;              // 0 or 1
    const int mt    = blockIdx.x * 8 + wave;

    if (mt * 16 < M) {                        // wave-uniform; EXEC stays full
        const float* __restrict__ Ap = A + (long long)(mt * 16 + l15) * K;
        const float* Bp = Bt + l15 * LDB + khalf * 2;

        v8f acc0 = {}, acc1 = {}, acc2 = {}, acc3 = {};

#pragma unroll
        for (int k0 = 0; k0 < K; k0 += 4) {
            v2f a = *(const v2f*)(Ap + k0 + khalf * 2);
            __builtin_prefetch(Ap + k0 + 16, 0, 1);     // global_prefetch_b8
            v2f b0 = *(const v2f*)(Bp + 0 * 16 * LDB + k0);
            v2f b1 = *(const v2f*)(Bp + 1 * 16 * LDB + k0);
            v2f b2 = *(const v2f*)(Bp + 2 * 16 * LDB + k0);
            v2f b3 = *(const v2f*)(Bp + 3 * 16 * LDB + k0);
            acc0 = __builtin_amdgcn_wmma_f32_16x16x4_f32(false, a, false, b0,
                                                         (short)0, acc0, false, false);
            acc1 = __builtin_amdgcn_wmma_f32_16x16x4_f32(false, a, false, b1,
                                                         (short)0, acc1, false, false);
            acc2 = __builtin_amdgcn_wmma_f32_16x16x4_f32(false, a, false, b2,
                                                         (short)0, acc2, false, false);
            acc3 = __builtin_amdgcn_wmma_f32_16x16x4_f32(false, a, false, b3,
                                                         (short)0, acc3, false, false);
        }

        // ---- epilogue ----
        const int crow0 = mt * 16 + khalf * 8;
        v8f accs[4] = {acc0, acc1, acc2, acc3};
#pragma unroll
        for (int i = 0; i < 4; ++i) {
            int col = n0 + i * 16 + l15;
            if (!NGUARD || col < N) {
                float bv = BIAS ? bias[col] : 0.0f;
#pragma unroll
                for (int r = 0; r < 8; ++r) {
                    C[(long long)(crow0 + r) * N + col] = accs[i][r] + bv;
                }
            }
        }
    }
}

// ---------------------------------------------------------------------------
// Host-side launcher
// ---------------------------------------------------------------------------
static inline unsigned nblk(long long n, int t) {
    return (unsigned)((n + t - 1) / t);
}

extern "C" void kernel_launch(void* const* d_in, const int* in_sizes, int n_in,
                              void* d_out, int out_size, void* d_ws, size_t ws_size,
                              hipStream_t stream) {
    const float*     x    = (const float*)d_in[0];
    const float*     W1   = (const float*)d_in[1];
    const float*     b1   = (const float*)d_in[2];
    const float*     W2   = (const float*)d_in[3];
    const float*     b2   = (const float*)d_in[4];
    const float*     Wfc  = (const float*)d_in[5];
    const float*     bfc  = (const float*)d_in[6];
    const long long* ei   = (const long long*)d_in[7];  // int64 edge_index [2,E]

    const int N = in_sizes[0] / DIN;   // nodes (100000, multiple of 16)
    const int E = in_sizes[7] / 2;
    const long long* src = ei;
    const long long* dst = ei + E;
    float* out = (float*)d_out;

    // workspace carve-up (floats)
    float* ws   = (float*)d_ws;
    float* dinv = ws;                 ws += N;                      // deg -> dinv in place
    float* H1   = ws;                 ws += (long long)N * HID;     // x @ W1
    float* A1   = ws;                 ws += (long long)N * HID;     // agg1 -> relu'd h1
    float* H2   = ws;                 ws += (long long)N * D2;      // h1 @ W2
    float* A2   = ws;                 ws += (long long)N * D2;      // agg2 -> relu'd h2

    // --- degree / normalization ---
    fill_zero<<<nblk(N, 256), 256, 0, stream>>>(dinv, N);
    deg_count<<<nblk(E, 256), 256, 0, stream>>>(dst, dinv, E);
    deg_to_dinv<<<nblk(N, 256), 256, 0, stream>>>(dinv, N);

    const unsigned mg = (unsigned)((N / 16 + 7) / 8);   // m-tile groups of 8

    // --- layer 1: H1 = x @ W1 ; scatter ; relu(agg + self + b1) ---
    wmma_gemm_f32<DIN, false, false><<<dim3(mg, HID / 64), 256, 0, stream>>>(
        x, W1, nullptr, H1, N, HID, 0);
    fill_zero<<<nblk((long long)N * HID, 256), 256, 0, stream>>>(A1, (long long)N * HID);
    edge_aggregate<<<nblk((long long)E * (HID / 4), 256), 256, 0, stream>>>(
        H1, src, dst, dinv, A1, E, HID);
    self_bias_relu<<<nblk((long long)N * HID, 256), 256, 0, stream>>>(
        A1, H1, dinv, b1, N, HID);

    // --- layer 2: H2 = A1 @ W2 ; scatter ; relu(agg + self + b2) ---
    wmma_gemm_f32<HID, false, false><<<dim3(mg, D2 / 64), 256, 0, stream>>>(
        A1, W2, nullptr, H2, N, D2, 0);
    fill_zero<<<nblk((long long)N * D2, 256), 256, 0, stream>>>(A2, (long long)N * D2);
    edge_aggregate<<<nblk((long long)E * (D2 / 4), 256), 256, 0, stream>>>(
        H2, src, dst, dinv, A2, E, D2);
    self_bias_relu<<<nblk((long long)N * D2, 256), 256, 0, stream>>>(
        A2, H2, dinv, b2, N, D2);

    // --- FC: out = A2 @ Wfc + bfc ---
    // full 64-col groups: cols [0, 960)
    wmma_gemm_f32<D2, true, false><<<dim3(mg, NOUT / 64), 256, 0, stream>>>(
        A2, Wfc, bfc, out, N, NOUT, 0);
    // guarded tail: cols [960, 1000)
    wmma_gemm_f32<D2, true, true><<<dim3(mg, 1), 256, 0, stream>>>(
        A2, Wfc, bfc, out, N, NOUT, (NOUT / 64) * 64);
}